// PlanLoss_45337674776962
// MI455X (gfx1250) — compile-verified
//
#include <hip/hip_runtime.h>
#include <math.h>

// ---------------------------------------------------------------------------
// PlanLoss for MI455X (gfx1250, wave32).
//   out = dist_loss + 2.5 * col_loss   (single f32 scalar)
// One block per n (512 blocks, 256 threads = 8 waves).
//   Phase 0: per-t box edges / valid mask / dist partials; per-(a,k)
//            {hw2, hh2, area1+area2+eps, v, v^2, 1+v+eps} with hoisted atans;
//            stage 64KB motion slab into padded LDS via async-to-LDS
//            (global_load_async_to_lds_b64 + s_wait_asynccnt).
//   Phase 1: wave w owns agents [16w, 16w+16). tc-outer / k-inner(unrolled):
//            each t-dependent LDS value loaded once per chunk; per-(a,k)
//            constants live in registers; divisions via v_rcp_f32. The
//            k-weighted CIoU sum goes into the f32 A-tile and one
//            V_WMMA_F32_16X16X4_F32 per 4-t chunk performs the valid-masked
//            time reduction on the matrix pipe (B = valid broadcast, D rows
//            replicated x16 across columns -> divide by 16 at the end).
//   Reductions: deterministic LDS trees; per-n partials in d_ws; final
//   single-block kernel combines.
// ---------------------------------------------------------------------------

typedef __attribute__((ext_vector_type(2))) float v2f;
typedef __attribute__((ext_vector_type(8))) float v8f;
typedef __attribute__((ext_vector_type(2))) int   v2i;

#define AS1 __attribute__((address_space(1)))
#define AS3 __attribute__((address_space(3)))

#define NN 512
#define AA 128
#define TT 64
#define KK 3
#define XPAD 65            // float2 row stride for xy2 slab (bank-conflict pad)
#define EPSF 1e-7f
#define FOUR_OVER_PI2 0.40528473456935108578f

#if defined(__has_builtin)
#if __has_builtin(__builtin_amdgcn_global_load_async_to_lds_b64) && \
    __has_builtin(__builtin_amdgcn_s_wait_asynccnt)
#define HAVE_ASYNC_LDS 1
#endif
#endif

__device__ __forceinline__ float fast_rcp(float x) {
    return __builtin_amdgcn_rcpf(x);   // v_rcp_f32, ~1 ulp — fine for a loss
}

__global__ __launch_bounds__(256) void plan_loss_main(
    const float* __restrict__ pred,    // [N][T][2]
    const float* __restrict__ targ,    // [N][T][2]
    const float* __restrict__ ego,     // [N][2]
    const float* __restrict__ msize,   // [N][A][2]
    const float* __restrict__ mmot,    // [N][A][T][2]
    float* __restrict__ ws)            // [3*N] : dist_num | valid_cnt | col_partial
{
    __shared__ float4 s_box1[TT];              // {l1, r1, t1, b1} per t
    __shared__ float2 s_c1[TT];                // {x1, y1} per t
    __shared__ __align__(8) float s_valid[TT]; // valid mask as float
    __shared__ float  s_dn[TT];                // dist numerator per t
    __shared__ float4 s_ak[AA * KK];           // {hw2, hh2, area2, v} per (a,k)
    __shared__ float2 s_xy2[AA * XPAD];        // padded agent-motion slab
    __shared__ float  s_red[256];
    __shared__ float  s_dist[2];               // {dist_num, valid_cnt}

    const int n   = blockIdx.x;
    const int tid = threadIdx.x;

    const float w1    = ego[n * 2 + 0];
    const float h1    = ego[n * 2 + 1];
    const float hw1   = 0.5f * w1;
    const float hh1   = 0.5f * h1;
    const float area1 = w1 * h1;
    const float atan1 = atanf(w1 / (h1 + EPSF));

    const float* slab = mmot + (size_t)n * (AA * TT * 2);

    // ---- Phase 0c(i): kick off motion-slab staging into padded LDS ----
#ifdef HAVE_ASYNC_LDS
    {
        const float2* slab2 = (const float2*)slab;
        for (int i = tid; i < AA * TT; i += 256) {
            const int a = i >> 6;
            const int t = i & 63;
            __builtin_amdgcn_global_load_async_to_lds_b64(
                (AS1 v2i*)(const void*)(slab2 + i),
                (AS3 v2i*)(void*)(&s_xy2[a * XPAD + t]),
                0, 0);
        }
    }
#else
    // Prefetch the 64KB slab (emits global_prefetch_b8), fill below.
    __builtin_prefetch(slab + tid * 16, 0, 1);
    __builtin_prefetch(slab + 4096 + tid * 16, 0, 1);
    __builtin_prefetch(slab + 8192 + tid * 16, 0, 1);
    __builtin_prefetch(slab + 12288 + tid * 16, 0, 1);
#endif

    // ---- Phase 0a: per-t quantities + dist partials (threads 0..63) ----
    if (tid < TT) {
        const float x1 = pred[(n * TT + tid) * 2 + 0];
        const float y1 = pred[(n * TT + tid) * 2 + 1];
        s_c1[tid]   = make_float2(x1, y1);
        s_box1[tid] = make_float4(x1 - hw1, x1 + hw1, y1 - hh1, y1 + hh1);

        const float tx = targ[(n * TT + tid) * 2 + 0];
        const float ty = targ[(n * TT + tid) * 2 + 1];
        const float vf = ((tx != -999.0f) && (ty != -999.0f)) ? 1.0f : 0.0f;
        s_valid[tid] = vf;
        const float dx = x1 - tx, dy = y1 - ty;
        s_dn[tid] = sqrtf(dx * dx + dy * dy) * vf;
    }

    // ---- Phase 0b: per-(a,k) constants with hoisted atan ----
    for (int i = tid; i < AA * KK; i += 256) {
        const int a = i / KK;
        const int k = i - a * KK;
        const float tilda = (k == 0) ? 0.0f : ((k == 1) ? 0.5f : 1.0f);
        const float w2 = msize[(n * AA + a) * 2 + 0] + tilda;
        const float h2 = msize[(n * AA + a) * 2 + 1] + tilda;
        const float da = atan1 - atanf(w2 * fast_rcp(h2 + EPSF));
        const float v  = FOUR_OVER_PI2 * da * da;
        s_ak[i] = make_float4(0.5f * w2, 0.5f * h2, w2 * h2, v);
    }

#ifdef HAVE_ASYNC_LDS
    __builtin_amdgcn_s_wait_asynccnt(0);
#else
    // ---- Phase 0c(ii): stage motion slab into padded LDS (coalesced b64) ----
    {
        const float2* slab2 = (const float2*)slab;
        for (int i = tid; i < AA * TT; i += 256) {
            const int a = i >> 6;
            const int t = i & 63;
            s_xy2[a * XPAD + t] = slab2[i];
        }
    }
#endif
    __syncthreads();

    if (tid == 0) {  // tiny serial reduction (deterministic)
        float dn = 0.0f, dd = 0.0f;
        for (int t = 0; t < TT; ++t) { dn += s_dn[t]; dd += s_valid[t]; }
        s_dist[0] = dn;
        s_dist[1] = dd;
    }

    // ---- Phase 1: CIoU + WMMA-accumulated masked t-reduction ----
    // 32-bit A-matrix layout (16x4): lanes 0-15: M=lane, VGPR0=K0, VGPR1=K1;
    // lanes 16-31: M=lane-16, VGPR0=K2, VGPR1=K3 -> lane computes
    // t = tc + 2*(lane>>4) + {0,1}. B[k][*] = valid[tc+k] broadcast.
    const int wave  = tid >> 5;
    const int lane  = tid & 31;
    const int row   = lane & 15;
    const int khalf = lane >> 4;
    const int a     = wave * 16 + row;

    // Hoist all per-(a,k) constants into registers.
    const float4 ak0 = s_ak[a * KK + 0];
    const float4 ak1 = s_ak[a * KK + 1];
    const float4 ak2 = s_ak[a * KK + 2];
    const float hw2k[KK] = {ak0.x, ak1.x, ak2.x};
    const float hh2k[KK] = {ak0.y, ak1.y, ak2.y};
    const float a12k[KK] = {area1 + ak0.z + EPSF,   // union+eps = a12 - inter
                            area1 + ak1.z + EPSF,
                            area1 + ak2.z + EPSF};
    const float vk2k[KK] = {ak0.w * ak0.w, ak1.w * ak1.w, ak2.w * ak2.w};
    const float opvk[KK] = {1.0f + ak0.w + EPSF,    // alpha_den = opv - iou
                            1.0f + ak1.w + EPSF,
                            1.0f + ak2.w + EPSF};
    const float wkk[KK]  = {1.0f, 0.4f, 0.1f};

    v8f acc = {0.f, 0.f, 0.f, 0.f, 0.f, 0.f, 0.f, 0.f};

    for (int tc = 0; tc < TT; tc += 4) {
        const int t0 = tc + khalf * 2;
        const float2 vld = *(const float2*)&s_valid[t0];   // t0 even -> aligned b64

        v2f av = {0.f, 0.f};
        #pragma unroll
        for (int sl = 0; sl < 2; ++sl) {
            const int t = t0 + sl;
            const float2 p2  = s_xy2[a * XPAD + t];
            const float4 b1v = s_box1[t];
            const float2 c1  = s_c1[t];

            const float dxc  = p2.x - c1.x;
            const float dyc  = p2.y - c1.y;
            const float rho2 = dxc * dxc + dyc * dyc;      // k-independent

            float wsum = 0.0f;
            #pragma unroll
            for (int k = 0; k < KK; ++k) {
                const float l2 = p2.x - hw2k[k], r2 = p2.x + hw2k[k];
                const float q2 = p2.y - hh2k[k], b2 = p2.y + hh2k[k];

                const float iw = fmaxf(fminf(b1v.y, r2) - fmaxf(b1v.x, l2), 0.0f);
                const float ih = fmaxf(fminf(b1v.w, b2) - fmaxf(b1v.z, q2), 0.0f);
                const float inter = iw * ih;
                const float iou   = inter * fast_rcp(a12k[k] - inter);

                const float cw = fmaxf(b1v.y, r2) - fminf(b1v.x, l2);
                const float ch = fmaxf(b1v.w, b2) - fminf(b1v.z, q2);
                const float c2 = fmaf(cw, cw, fmaf(ch, ch, EPSF));

                const float ciou = iou - rho2 * fast_rcp(c2)
                                       - vk2k[k] * fast_rcp(opvk[k] - iou);
                wsum = fmaf(wkk[k], ciou, wsum);
            }
            av[sl] = wsum;
        }
        const v2f bv = {vld.x, vld.y};
        // D += A(16x4 weighted-ciou) * B(4x16 valid broadcast) — matrix pipe
        acc = __builtin_amdgcn_wmma_f32_16x16x4_f32(
            false, av, false, bv, (short)0, acc, false, false);
    }

    const float mysum = acc[0] + acc[1] + acc[2] + acc[3] +
                        acc[4] + acc[5] + acc[6] + acc[7];
    s_red[tid] = mysum;
    __syncthreads();
    #pragma unroll
    for (int off = 128; off > 0; off >>= 1) {
        if (tid < off) s_red[tid] += s_red[tid + off];
        __syncthreads();
    }

    if (tid == 0) {
        ws[n]           = s_dist[0];                               // dist numerator
        ws[NN + n]      = s_dist[1];                               // valid count
        ws[2 * NN + n]  = (s_red[0] * (1.0f / 16.0f)) / s_dist[1]; // sum_a col
    }
}

__global__ __launch_bounds__(512) void plan_loss_final(
    const float* __restrict__ ws, float* __restrict__ out)
{
    __shared__ float r0[NN], r1[NN], r2[NN];
    const int tid = threadIdx.x;
    r0[tid] = ws[tid];
    r1[tid] = ws[NN + tid];
    r2[tid] = ws[2 * NN + tid];
    __syncthreads();
    #pragma unroll
    for (int off = NN / 2; off > 0; off >>= 1) {
        if (tid < off) {
            r0[tid] += r0[tid + off];
            r1[tid] += r1[tid + off];
            r2[tid] += r2[tid + off];
        }
        __syncthreads();
    }
    if (tid == 0) {
        const float dist_loss = r0[0] / r1[0];
        const float col_loss  = r2[0] / (float)(NN * AA);
        out[0] = 1.0f * dist_loss + 2.5f * col_loss;
    }
}

extern "C" void kernel_launch(void* const* d_in, const int* in_sizes, int n_in,
                              void* d_out, int out_size, void* d_ws, size_t ws_size,
                              hipStream_t stream) {
    (void)in_sizes; (void)n_in; (void)out_size; (void)ws_size;
    const float* pred  = (const float*)d_in[0];  // [512][64][2]
    const float* targ  = (const float*)d_in[1];  // [512][64][2]
    const float* ego   = (const float*)d_in[2];  // [512][2]
    const float* msize = (const float*)d_in[3];  // [512][128][2]
    const float* mmot  = (const float*)d_in[4];  // [512][128][64][2]
    float* ws  = (float*)d_ws;                   // 3*512 floats
    float* out = (float*)d_out;                  // 1 float

    plan_loss_main<<<NN, 256, 0, stream>>>(pred, targ, ego, msize, mmot, ws);
    plan_loss_final<<<1, NN, 0, stream>>>(ws, out);
}